// YOLOLoss_23252952940853
// MI455X (gfx1250) — compile-verified
//
#include <hip/hip_runtime.h>
#include <hip/hip_bf16.h>
#include <math.h>

// ---------------- problem constants (from reference) ----------------
#define BB 16
#define AA 3
#define HH 80
#define WW 80
#define CC 84            // 4 box + 80 classes
#define HWC (HH*WW)      // 6400
#define NTGT 64
#define NROWS (BB*AA*NTGT)        // 3072 gathered rows
#define NCELLS (BB*AA*HWC)        // 307200 objectness cells
#define BDIM 256

typedef __attribute__((ext_vector_type(2))) float v2f;
typedef __attribute__((ext_vector_type(8))) float v8f;

// numerically stable softplus, matches jax.nn.softplus semantics
__device__ __forceinline__ float softplus_f(float x) {
  return fmaxf(x, 0.0f) + log1pf(expf(-fabsf(x)));
}

// Full-wave (32 lane) fp32 sum. Uses V_WMMA_F32_16X16X4_F32 with B = ones:
// A[m,k] layout: lanes 0-15 -> VGPR0=K0, lanes 16-31 -> VGPR0=K2 (VGPR1 zeroed),
// so D[m,n] = p[m] + p[m+16]. Summing the 8 D VGPRs gives the half-row-set sum;
// one shfl_xor(16) completes the reduction. fp32 throughout (no precision loss).
__device__ __forceinline__ float wave_sum(float x) {
#if __has_builtin(__builtin_amdgcn_wmma_f32_16x16x4_f32)
  v2f a; a[0] = x;    a[1] = 0.0f;
  v2f b; b[0] = 1.0f; b[1] = 1.0f;
  v8f c = {0.f, 0.f, 0.f, 0.f, 0.f, 0.f, 0.f, 0.f};
  v8f d = __builtin_amdgcn_wmma_f32_16x16x4_f32(
      /*neg_a=*/false, a, /*neg_b=*/false, b,
      /*c_mod=*/(short)0, c, /*reuse_a=*/false, /*reuse_b=*/false);
  float s = d[0] + d[1] + d[2] + d[3] + d[4] + d[5] + d[6] + d[7];
  s += __shfl_xor(s, 16, 32);
  return s;
#else
  for (int off = 16; off > 0; off >>= 1) x += __shfl_xor(x, off, 32);
  return x;
#endif
}

// block reduce (8 waves) + single atomic per block
__device__ __forceinline__ void block_reduce_atomic(float part, float* out, float* sm) {
  float w = wave_sum(part);               // EXEC all-1s here: no divergence at call site
  const int lane = threadIdx.x & 31;
  const int wid  = threadIdx.x >> 5;
  if (lane == 0) sm[wid] = w;
  __syncthreads();
  if (threadIdx.x == 0) {
    float t = 0.0f;
    #pragma unroll
    for (int i = 0; i < BDIM / 32; ++i) t += sm[i];
    atomicAdd(out, t);                    // fire-and-forget f32 atomic (no return)
  }
}

// ---------------- kernel 1: per-target prep + zero output ----------------
__global__ void yolo_prep(const float* __restrict__ tgt, float* __restrict__ out,
                          int* __restrict__ cells, int* __restrict__ clsid,
                          int* __restrict__ firstocc, float* __restrict__ box,
                          float* __restrict__ wno) {
  const int t = threadIdx.x;
  if (t < NTGT) {
    const float cid = tgt[t * 5 + 0];
    const float tx  = tgt[t * 5 + 1];
    const float ty  = tgt[t * 5 + 2];
    int gx = (int)floorf(tx * (float)WW); gx = min(max(gx, 0), WW - 1);
    int gy = (int)floorf(ty * (float)HH); gy = min(max(gy, 0), HH - 1);
    cells[t] = gy * WW + gx;
    clsid[t] = (int)cid;
    box[t * 4 + 0] = tx;
    box[t * 4 + 1] = ty;
    box[t * 4 + 2] = tgt[t * 5 + 3];
    box[t * 4 + 3] = tgt[t * 5 + 4];
  }
  __syncthreads();
  if (t == 0) {
    int nobj = 0;                         // dedup cells (reference uses a 0/1 mask)
    for (int i = 0; i < NTGT; ++i) {
      int f = 1;
      for (int j = 0; j < i; ++j)
        if (cells[j] == cells[i]) { f = 0; break; }
      firstocc[i] = f;
      nobj += f;
    }
    const float noobj_count = (float)(BB * AA) * ((float)HWC - (float)nobj);
    *wno = 0.5f / noobj_count;            // noobj_scale / noobj_count
    *out = 0.0f;                          // accumulators target d_out directly
  }
}

// ---------------- kernel 2: streaming softplus over channel 4 ----------------
// noobj term = wno * [ sum_all sp(ch4) - sum_objcells sp(ch4) ]; this kernel is
// the +sum_all part. Stride-336B stream: one cacheline per element (~39MB).
__global__ void yolo_noobj(const float* __restrict__ pred, float* __restrict__ out,
                           const float* __restrict__ wno_p) {
  __shared__ float sm[BDIM / 32];
  const float wno = *wno_p;
  const int tid    = blockIdx.x * blockDim.x + threadIdx.x;
  const int stride = gridDim.x * blockDim.x;
  float part = 0.0f;
  for (int i = tid; i < NCELLS; i += stride) {
    if (i + stride < NCELLS)              // WGP-scope prefetch is non-speculative: keep in-bounds
      __builtin_prefetch(&pred[(size_t)(i + stride) * CC + 4], 0, 0);
    part += softplus_f(pred[(size_t)i * CC + 4]);
  }
  block_reduce_atomic(part * wno, out, sm);
}

// ---------------- kernel 3: gathered rows (loc + cls + obj - noobj@objcells) ----------------
__global__ void yolo_obj(const float* __restrict__ pred, float* __restrict__ out,
                         const int* __restrict__ cells, const int* __restrict__ clsid,
                         const int* __restrict__ firstocc, const float* __restrict__ box,
                         const float* __restrict__ wno_p) {
  __shared__ float sm[BDIM / 32];
  const float w_loc = 1.0f / (float)(NROWS * 4);   // mean over (B,A,64,4)
  const float w_cls = 1.0f / (float)(NROWS * 80);  // mean over (B,A,64,80)
  const float w_obj = 1.0f / (float)(NROWS);       // mean over (B,A,64)
  const float wno   = *wno_p;

  const int lane   = threadIdx.x & 31;
  const int gwid   = (blockIdx.x * blockDim.x + threadIdx.x) >> 5;
  const int nwaves = (gridDim.x * blockDim.x) >> 5;

  float part = 0.0f;
  for (int r = gwid; r < NROWS; r += nwaves) {     // one wave per gathered row
    const int t  = r & (NTGT - 1);
    const int ba = r >> 6;
    const float* __restrict__ row = pred + ((size_t)ba * HWC + (size_t)cells[t]) * CC;
    const int cid = clsid[t];
    const int fo  = firstocc[t];
    for (int c = lane; c < CC; c += 32) {
      const float p = row[c];
      if (c < 4) {
        const float d = p - box[t * 4 + c];
        part += w_loc * d * d;
      } else {
        const float s = softplus_f(p);
        // logits = pc[...,4:], class channel index = c-4
        part += w_cls * (s - ((c - 4) == cid ? p : 0.0f));
        if (c == 4) {
          part += w_obj * softplus_f(-p);          // obj term: softplus(-x)
          if (fo) part -= wno * s;                 // subtract obj-cell from noobj sum
        }
      }
    }
  }
  block_reduce_atomic(part, out, sm);
}

// ---------------- launch ----------------
extern "C" void kernel_launch(void* const* d_in, const int* in_sizes, int n_in,
                              void* d_out, int out_size, void* d_ws, size_t ws_size,
                              hipStream_t stream) {
  (void)in_sizes; (void)n_in; (void)out_size; (void)ws_size;
  const float* pred = (const float*)d_in[0];
  const float* tgt  = (const float*)d_in[1];
  float* out = (float*)d_out;

  // workspace layout (all rewritten by yolo_prep every call)
  int*   cells    = (int*)d_ws;                    // 64 ints
  int*   clsid    = cells + NTGT;                  // 64 ints
  int*   firstocc = clsid + NTGT;                  // 64 ints
  float* box      = (float*)(firstocc + NTGT);     // 256 floats
  float* wno      = box + NTGT * 4;                // 1 float

  yolo_prep<<<1, 64, 0, stream>>>(tgt, out, cells, clsid, firstocc, box, wno);
  yolo_noobj<<<512, BDIM, 0, stream>>>(pred, out, wno);   // 131072 threads, 2-3 elems each
  yolo_obj<<<64, BDIM, 0, stream>>>(pred, out, cells, clsid, firstocc, box, wno); // 512 waves, 6 rows each
}